// HCHA_41566693490885
// MI455X (gfx1250) — compile-verified
//
#include <hip/hip_runtime.h>
#include <hip/hip_bf16.h>

#define N_NODES  50000
#define N_HEDGES 10000
#define NNZ      640000
#define DF       128            // feature width (D_IN == D_HID == D_OUT)

typedef __attribute__((ext_vector_type(2))) float v2f;
typedef __attribute__((ext_vector_type(8))) float v8f;

// ---------------------------------------------------------------------------
// Dense GEMM  Y[nrows x 128] = X[nrows x 128] * W[128 x 128]  (fp32 WMMA)
// One wave per 16x16 output tile; block = 256 threads = 8 waves covering
// 16 rows x 128 cols. K-loop: 32 x V_WMMA_F32_16X16X4_F32, fully unrolled.
// ---------------------------------------------------------------------------
__global__ __launch_bounds__(256) void hcha_gemm_wmma_f32(
    const float* __restrict__ X, const float* __restrict__ W,
    float* __restrict__ Y, int nrows)
{
    const int wave = threadIdx.x >> 5;
    const int lane = threadIdx.x & 31;
    const int row0 = blockIdx.x * 16;
    const int col0 = wave * 16;
    if (row0 >= nrows) return;

    const int mn   = lane & 15;          // M index for A, N index for B/C/D
    const int koff = (lane >> 4) << 1;   // lanes 0-15 -> K{0,1}; 16-31 -> K{2,3}

    const float* __restrict__ xrow = X + (size_t)(row0 + mn) * DF;

    v8f acc = {0.f, 0.f, 0.f, 0.f, 0.f, 0.f, 0.f, 0.f};
#pragma unroll
    for (int k = 0; k < DF; k += 4) {
        v2f a, b;
        a.x = xrow[k + koff];
        a.y = xrow[k + koff + 1];
        b.x = W[(size_t)(k + koff)     * DF + col0 + mn];
        b.y = W[(size_t)(k + koff + 1) * DF + col0 + mn];
        // (neg_a, A, neg_b, B, c_mod, C, reuse_a, reuse_b)
        acc = __builtin_amdgcn_wmma_f32_16x16x4_f32(
            false, a, false, b, (short)0, acc, false, false);
    }

    // C/D layout: VGPR v -> row (v + 8*(lane>=16)), col = lane&15
    const int crow = (lane >> 4) * 8;
    float* __restrict__ yout = Y + (size_t)row0 * DF + col0;
#pragma unroll
    for (int v = 0; v < 8; ++v)
        yout[(size_t)(crow + v) * DF + mn] = acc[v];
}

// ---------------------------------------------------------------------------
// Degree counting: one thread per incidence.
// ---------------------------------------------------------------------------
__global__ __launch_bounds__(256) void hcha_degree(
    const long long* __restrict__ row, const long long* __restrict__ col,
    float* __restrict__ deg_n, float* __restrict__ deg_e, int nnz)
{
    int i = blockIdx.x * blockDim.x + threadIdx.x;
    if (i < nnz) {
        atomicAdd(deg_n + row[i], 1.0f);
        atomicAdd(deg_e + col[i], 1.0f);
    }
}

__global__ __launch_bounds__(256) void hcha_inv(float* __restrict__ d, int n)
{
    int i = blockIdx.x * blockDim.x + threadIdx.x;
    if (i < n) {
        float v = d[i];
        d[i] = (v > 0.f) ? (1.0f / v) : 0.f;
    }
}

// ---------------------------------------------------------------------------
// Segment-sum scatter: one wave per incidence; lane handles 4 floats.
// dst[dst_idx[e]][:] += src[src_idx[e]][:]   (fp32 atomics resolve in L2)
// ---------------------------------------------------------------------------
__global__ __launch_bounds__(256) void hcha_scatter128(
    const float* __restrict__ src, float* __restrict__ dst,
    const long long* __restrict__ src_idx, const long long* __restrict__ dst_idx,
    int nnz)
{
    const int wave = blockIdx.x * (blockDim.x >> 5) + (threadIdx.x >> 5);
    const int lane = threadIdx.x & 31;
    if (wave >= nnz) return;

    const long long s = src_idx[wave];
    const long long d = dst_idx[wave];

    const float4 v = ((const float4*)(src + (size_t)s * DF))[lane];
    float* __restrict__ dp = dst + (size_t)d * DF + lane * 4;
    atomicAdd(dp + 0, v.x);
    atomicAdd(dp + 1, v.y);
    atomicAdd(dp + 2, v.z);
    atomicAdd(dp + 3, v.w);
}

// mat[r][c] *= inv[r]
__global__ __launch_bounds__(256) void hcha_scale_rows(
    float* __restrict__ mat, const float* __restrict__ inv, int rows)
{
    size_t idx = (size_t)blockIdx.x * blockDim.x + threadIdx.x;
    int r = (int)(idx >> 7);
    if (r < rows) mat[idx] *= inv[r];
}

// out[r][c] = maybe_relu(acc[r][c]*inv[r] + bias[c])   (in-place safe)
__global__ __launch_bounds__(256) void hcha_finalize(
    const float* __restrict__ acc, const float* __restrict__ inv,
    const float* __restrict__ bias, float* __restrict__ out,
    int rows, int do_relu)
{
    size_t idx = (size_t)blockIdx.x * blockDim.x + threadIdx.x;
    int r = (int)(idx >> 7);
    int c = (int)(idx & (DF - 1));
    if (r < rows) {
        float v = fmaf(acc[idx], inv[r], bias[c]);
        out[idx] = do_relu ? fmaxf(v, 0.f) : v;
    }
}

// ---------------------------------------------------------------------------
extern "C" void kernel_launch(void* const* d_in, const int* in_sizes, int n_in,
                              void* d_out, int out_size, void* d_ws, size_t ws_size,
                              hipStream_t stream)
{
    const float*     x   = (const float*)d_in[0];
    const long long* ei  = (const long long*)d_in[1];   // int64 [2, NNZ]
    const float*     W1  = (const float*)d_in[2];
    const float*     b1  = (const float*)d_in[3];
    const float*     W2  = (const float*)d_in[4];
    const float*     b2  = (const float*)d_in[5];
    const long long* row = ei;            // node index per incidence
    const long long* col = ei + NNZ;      // hyperedge index per incidence
    float* out = (float*)d_out;

    const size_t XW_BYTES = (size_t)N_NODES  * DF * sizeof(float);   // 25.6 MB
    const size_t M_BYTES  = (size_t)N_HEDGES * DF * sizeof(float);   //  5.1 MB

    char*  ws = (char*)d_ws;
    float* xw = (float*)ws;                                 // [N, 128]
    float* h  = (float*)(ws + XW_BYTES);                    // [N, 128]
    float* m  = (float*)(ws + 2 * XW_BYTES);                // [E, 128]
    float* dn = (float*)(ws + 2 * XW_BYTES + M_BYTES);      // [N] 1/deg
    float* de = dn + N_NODES;                               // [E] 1/deg

    const int T = 256;
    const dim3 gScat(NNZ / 8);                 // 8 waves/block, 1 edge/wave
    const dim3 gGemm(N_NODES / 16);            // 50000 = 3125 * 16
    const dim3 gNodeEl((N_NODES * DF) / T);    // elementwise over [N,128]
    const dim3 gEdgeEl((N_HEDGES * DF) / T);   // elementwise over [E,128]

    // ---- degrees ---------------------------------------------------------
    hipMemsetAsync(dn, 0, (size_t)(N_NODES + N_HEDGES) * sizeof(float), stream);
    hipMemsetAsync(m, 0, M_BYTES, stream);
    hipMemsetAsync(h, 0, XW_BYTES, stream);
    hcha_degree<<<(NNZ + T - 1) / T, T, 0, stream>>>(row, col, dn, de, NNZ);
    hcha_inv<<<(N_NODES + T - 1) / T, T, 0, stream>>>(dn, N_NODES);
    hcha_inv<<<(N_HEDGES + T - 1) / T, T, 0, stream>>>(de, N_HEDGES);

    // ---- layer 1: h = relu(D^-1 H B^-1 H^T (x W1) + b1) ------------------
    hcha_gemm_wmma_f32<<<gGemm, T, 0, stream>>>(x, W1, xw, N_NODES);
    hcha_scatter128<<<gScat, T, 0, stream>>>(xw, m, row, col, NNZ);   // node->edge
    hcha_scale_rows<<<gEdgeEl, T, 0, stream>>>(m, de, N_HEDGES);      // * B^-1
    hcha_scatter128<<<gScat, T, 0, stream>>>(m, h, col, row, NNZ);    // edge->node
    hcha_finalize<<<gNodeEl, T, 0, stream>>>(h, dn, b1, h, N_NODES, 1);

    // ---- layer 2: out = D^-1 H B^-1 H^T (h W2) + b2 ----------------------
    hcha_gemm_wmma_f32<<<gGemm, T, 0, stream>>>(h, W2, xw, N_NODES);
    hipMemsetAsync(m, 0, M_BYTES, stream);
    hcha_scatter128<<<gScat, T, 0, stream>>>(xw, m, row, col, NNZ);
    hcha_scale_rows<<<gEdgeEl, T, 0, stream>>>(m, de, N_HEDGES);
    hipMemsetAsync(out, 0, XW_BYTES, stream);
    hcha_scatter128<<<gScat, T, 0, stream>>>(m, out, col, row, NNZ);
    hcha_finalize<<<gNodeEl, T, 0, stream>>>(out, dn, b2, out, N_NODES, 0);
}